// BASE_1692217115458
// MI455X (gfx1250) — compile-verified
//
#include <hip/hip_runtime.h>
#include <math.h>

// ---- problem constants (fixed by reference) ----
#define NGRAPHS 64
#define NPG     1024
#define NFEAT   512
#define EDEG    32
#define DEF     16                       // edge-feature dim
#define NNODES  (NGRAPHS * NPG)          // 65536
#define NEDGES  (NNODES * EDEG)          // 2097152
#define KSEL    (NPG / 2)                // 512
#define NSEL    (NGRAPHS * KSEL)         // 32768

// ---- d_out layout: outputs concatenated flat in return order, as float ----
#define OFF_XNEW  ((size_t)0)
#define OFF_EI    ((size_t)NSEL * NFEAT)                 // 16777216
#define OFF_EA    (OFF_EI + (size_t)2 * NEDGES)          // 20971520
#define OFF_MASK  (OFF_EA + (size_t)NEDGES * DEF)        // 54525952
#define OFF_BATCH (OFF_MASK + (size_t)NEDGES)            // 56623104
#define OFF_PERM  (OFF_BATCH + (size_t)NSEL)             // 56655872

typedef __attribute__((ext_vector_type(2))) float v2f;
typedef __attribute__((ext_vector_type(8))) float v8f;

// ---------------- degree ----------------
__global__ void k_init_deg(float* __restrict__ deg) {
  int i = blockIdx.x * blockDim.x + threadIdx.x;
  if (i < NNODES) deg[i] = 1.0f;  // self-loop
}

__global__ void k_deg_scatter(const int* __restrict__ ei, float* __restrict__ deg) {
  int e = blockIdx.x * blockDim.x + threadIdx.x;
  if (e < NEDGES) atomicAdd(&deg[ei[NEDGES + e]], 1.0f);
}

// ---------------- GEMV h = x @ W via fp32 WMMA (16x16x4) ----------------
// One wave32 computes 16 rows. A-layout (32-bit A 16x4): lane L holds row
// M = L&15; VGPR0/1 hold K-pair 2*(L>>4) + {0,1}. B columns all carry W so
// every D column equals the score; extract column 0 from lanes 0 and 16.
__global__ void k_gemv_wmma(const float* __restrict__ x,
                            const float* __restrict__ W,
                            float* __restrict__ h) {
  int lane = threadIdx.x & 31;
  int wave = (blockIdx.x * blockDim.x + threadIdx.x) >> 5;
  int m0   = wave << 4;                       // grid sized exactly: no bounds check
  int row  = m0 + (lane & 15);
  int kh   = (lane >> 4) << 1;                // 0 for lanes 0-15, 2 for lanes 16-31
  const float* xr = x + (size_t)row * NFEAT;
  v8f acc = {};
  for (int kk = 0; kk < NFEAT; kk += 4) {
    v2f a;  a.x  = xr[kk + kh];  a.y  = xr[kk + kh + 1];
    v2f bm; bm.x = W[kk + kh];   bm.y = W[kk + kh + 1];
    acc = __builtin_amdgcn_wmma_f32_16x16x4_f32(
        /*neg_a=*/false, a, /*neg_b=*/false, bm,
        /*c_mod=*/(short)0, acc, /*reuse_a=*/false, /*reuse_b=*/false);
  }
  // D (16x16 f32): VGPR r -> M=r (lanes 0-15, N=lane), M=r+8 (lanes 16-31)
  if ((lane & 15) == 0) {
    int base = m0 + ((lane >> 4) << 3);
    #pragma unroll
    for (int r = 0; r < 8; ++r) h[base + r] = acc[r];
  }
}

// ---------------- score = self-loop + bias, then edge scatter ----------------
__global__ void k_score_init(const float* __restrict__ deg, const float* __restrict__ h,
                             const float* __restrict__ b, float* __restrict__ dinv,
                             float* __restrict__ score) {
  int i = blockIdx.x * blockDim.x + threadIdx.x;
  if (i < NNODES) {
    float di = rsqrtf(deg[i]);
    dinv[i]  = di;
    score[i] = h[i] * di * di + b[0];
  }
}

__global__ void k_score_scatter(const int* __restrict__ ei, const float* __restrict__ h,
                                const float* __restrict__ dinv, float* __restrict__ score) {
  int e = blockIdx.x * blockDim.x + threadIdx.x;
  if (e < NEDGES) {
    int s = ei[e], d = ei[NEDGES + e];
    atomicAdd(&score[d], h[s] * dinv[s] * dinv[d]);
  }
}

// ---------------- per-graph top-k via in-LDS bitonic sort (descending) ----------------
__global__ __launch_bounds__(NPG) void k_topk(const float* __restrict__ score,
                                              int* __restrict__ perm,
                                              int* __restrict__ newidx) {
  __shared__ float ss[NPG];
  __shared__ int   si[NPG];
  int t    = threadIdx.x;
  int base = blockIdx.x * NPG;
  ss[t] = score[base + t];
  si[t] = t;
  newidx[base + t] = -1;                       // this graph's nodes only
  __syncthreads();
  for (int ksz = 2; ksz <= NPG; ksz <<= 1) {
    for (int j = ksz >> 1; j > 0; j >>= 1) {
      int p = t ^ j;
      if (p > t) {
        float a = ss[t], c = ss[p];
        int  ia = si[t], ic = si[p];
        // precedence in descending order (index tie-break ascending, top_k-like)
        bool aFirst = (a > c) || (a == c && ia < ic);
        bool desc   = ((t & ksz) == 0);
        if (desc ? !aFirst : aFirst) { ss[t] = c; ss[p] = a; si[t] = ic; si[p] = ia; }
      }
      __syncthreads();
    }
  }
  if (t < KSEL) {
    int node   = base + si[t];
    int outpos = blockIdx.x * KSEL + t;
    perm[outpos] = node;
    newidx[node] = outpos;
  }
}

// ---------------- x_new = x[perm] * tanh(score[perm]); batch_new; perm ----------------
__global__ void k_gather(const float* __restrict__ x, const float* __restrict__ score,
                         const int* __restrict__ perm, float* __restrict__ out) {
  int rowo = blockIdx.x;                       // [0, NSEL)
  int src  = perm[rowo];
  float scale = tanhf(score[src]);
  const float4* xin = (const float4*)(x + (size_t)src * NFEAT);
  float4*       xo  = (float4*)(out + OFF_XNEW + (size_t)rowo * NFEAT);
  int t = threadIdx.x;                         // 128 threads x float4
  float4 v = xin[t];
  v.x *= scale; v.y *= scale; v.z *= scale; v.w *= scale;
  xo[t] = v;
  if (t == 0) {
    out[OFF_BATCH + rowo] = (float)(src >> 10);   // batch id = node / NPG
    out[OFF_PERM  + rowo] = (float)src;
  }
}

// ---------------- edge filter + remap + masked edge_attr ----------------
__global__ void k_edges(const int* __restrict__ ei, const float* __restrict__ ea,
                        const int* __restrict__ newidx, float* __restrict__ out) {
  int e = blockIdx.x * blockDim.x + threadIdx.x;
  if (e >= NEDGES) return;
  int s = ei[e], d = ei[NEDGES + e];
  int ns = newidx[s], nd = newidx[d];
  bool m = (ns >= 0) && (nd >= 0);
  out[OFF_EI + e]              = (float)(m ? ns : -1);
  out[OFF_EI + NEDGES + e]     = (float)(m ? nd : -1);
  out[OFF_MASK + e]            = m ? 1.0f : 0.0f;
  const float4* eap = (const float4*)(ea + (size_t)e * DEF);
  float4*       eao = (float4*)(out + OFF_EA + (size_t)e * DEF);
  #pragma unroll
  for (int c = 0; c < DEF / 4; ++c) {
    float4 v = eap[c];
    if (!m) { v.x = 0.f; v.y = 0.f; v.z = 0.f; v.w = 0.f; }
    eao[c] = v;
  }
}

extern "C" void kernel_launch(void* const* d_in, const int* in_sizes, int n_in,
                              void* d_out, int out_size, void* d_ws, size_t ws_size,
                              hipStream_t stream) {
  (void)in_sizes; (void)n_in; (void)out_size; (void)ws_size;
  const float* x  = (const float*)d_in[0];
  const int*   ei = (const int*)d_in[1];     // [2, E] row-major
  const float* ea = (const float*)d_in[2];
  const float* W  = (const float*)d_in[4];
  const float* b  = (const float*)d_in[5];
  float* out = (float*)d_out;

  // workspace carve-out (~1.4 MB)
  float* deg   = (float*)d_ws;
  float* h     = deg  + NNODES;
  float* dinv  = h    + NNODES;
  float* score = dinv + NNODES;
  int*   perm   = (int*)(score + NNODES);
  int*   newidx = perm + NSEL;

  k_init_deg     <<<NNODES / 256, 256, 0, stream>>>(deg);
  k_deg_scatter  <<<NEDGES / 256, 256, 0, stream>>>(ei, deg);
  k_gemv_wmma    <<<(NNODES / 16) * 32 / 256, 256, 0, stream>>>(x, W, h);
  k_score_init   <<<NNODES / 256, 256, 0, stream>>>(deg, h, b, dinv, score);
  k_score_scatter<<<NEDGES / 256, 256, 0, stream>>>(ei, h, dinv, score);
  k_topk         <<<NGRAPHS, NPG, 0, stream>>>(score, perm, newidx);
  k_gather       <<<NSEL, NFEAT / 4, 0, stream>>>(x, score, perm, out);
  k_edges        <<<NEDGES / 256, 256, 0, stream>>>(ei, ea, newidx, out);
}